// HeatMapParser_71536975282595
// MI455X (gfx1250) — compile-verified
//
#include <hip/hip_runtime.h>
#include <hip/hip_bf16.h>
#include <stdint.h>

// ---------------------------------------------------------------------------
// HeatMapParser for MI455X (gfx1250).
// 0 FLOPs, ~0.54 GB of movement -> HBM-bandwidth bound (~23us @ 23.3 TB/s).
// CDNA5 path: Tensor Data Mover (TENSOR_LOAD_TO_LDS / TENSOR_STORE_FROM_LDS,
// TENSORcnt) for the strided 64x64 patch gather; ASYNCcnt global<->LDS DMA
// kernel retained as alternate path.
// ---------------------------------------------------------------------------

#define BS 4
#define C  64
#define H  512
#define W  512
#define CH 64
#define CW 64
#define RY 8
#define RX 8
#define NREG (BS * RY * RX)                        // 256 regions
#define PATCH_ELEMS ((size_t)NREG * C * CH * CW)   // 67,108,864 floats

typedef unsigned int u32x4 __attribute__((ext_vector_type(4)));
typedef unsigned int u32x8 __attribute__((ext_vector_type(8)));

__device__ __forceinline__ int iabs_(int v) { return v < 0 ? -v : v; }

// ----- gfx1250 async LDS<->global DMA (ASYNCcnt) ---------------------------
__device__ __forceinline__ void async_load_b32(uint32_t lds_addr, const float* gaddr) {
    asm volatile("global_load_async_to_lds_b32 %0, %1, off"
                 :: "v"(lds_addr), "v"((unsigned long long)(uintptr_t)gaddr)
                 : "memory");
}
__device__ __forceinline__ void async_store_b32(float* gaddr, uint32_t lds_addr) {
    asm volatile("global_store_async_from_lds_b32 %0, %1, off"
                 :: "v"((unsigned long long)(uintptr_t)gaddr), "v"(lds_addr)
                 : "memory");
}
__device__ __forceinline__ void wait_async0() {
    asm volatile("s_wait_asynccnt 0" ::: "memory");
}

// ----- gfx1250 Tensor Data Mover (TENSORcnt) -------------------------------
// D# group0 (4 SGPRs) + group1 (8 SGPRs); VADDR2/3 omitted (NULL) => 2D tensor.
__device__ __forceinline__ void tdm_load(u32x4 g0, u32x8 g1) {
    asm volatile("tensor_load_to_lds %0, %1" :: "s"(g0), "s"(g1) : "memory");
}
__device__ __forceinline__ void tdm_store(u32x4 g0, u32x8 g1) {
    asm volatile("tensor_store_from_lds %0, %1" :: "s"(g0), "s"(g1) : "memory");
}

// Build D# group0: count=1, lds byte address, 57b global byte address, type=2.
__device__ __forceinline__ u32x4 make_g0(uint32_t lds_addr, uint64_t gaddr) {
    u32x4 g;
    g[0] = 1u;                                           // count=1, user mode
    g[1] = lds_addr;                                     // lds_addr [63:32]
    g[2] = (uint32_t)gaddr;                              // global_addr lo
    g[3] = ((uint32_t)(gaddr >> 32) & 0x01FFFFFFu)       // global_addr [56:32]
         | 0x80000000u;                                  // type=2 ("image")
    return g;
}

// Build D# group1 for a 2D tile of 4-byte elements.
// tensor_dim0/1: tensor extent (elements); tile_dim0/1: tile extent;
// stride0: elements between rows.
__device__ __forceinline__ u32x8 make_g1(uint32_t tdim0, uint32_t tdim1,
                                         uint32_t tile0, uint32_t tile1,
                                         uint32_t stride0) {
    u32x8 g;
    g[0] = 0x00020000u;                  // workgroup_mask=0, data_size=2 (4B)
    g[1] = (tdim0 & 0xFFFFu) << 16;      // tensor_dim0[15:0] (barrier_addr=0)
    g[2] = ((tdim0 >> 16) & 0xFFFFu) | ((tdim1 & 0xFFFFu) << 16);
    g[3] = ((tdim1 >> 16) & 0xFFFFu) | ((tile0 & 0xFFFFu) << 16);
    g[4] = (tile1 & 0xFFFFu);            // tile_dim1; tile_dim2=0
    g[5] = stride0;                      // tensor_dim0_stride lo32
    g[6] = 0u;                           // stride0 hi16 | stride1 lo16
    g[7] = 0u;                           // stride1 hi
    return g;
}

// ---------------------------------------------------------------------------
// Stage 1: heatmap analysis (unchanged from round 1; compiles clean).
// ---------------------------------------------------------------------------
__global__ void heatmap_stage1(const float* __restrict__ mask,
                               float* __restrict__ offsets_out,  // NREG*5 floats
                               float* __restrict__ valid_out,    // NREG floats
                               int* __restrict__ ws)             // NREG*3 ints
{
    __shared__ unsigned actCol[CW];
    __shared__ unsigned actRow[CH];
    __shared__ unsigned anyCen;

    const int n  = blockIdx.x;
    const int b  = n >> 6;
    const int rm = n & 63;
    const int gy = rm >> 3;
    const int gx = rm & 7;
    const int t  = threadIdx.x;      // 256 threads

    if (t < CW) { actCol[t] = 0u; actRow[t] = 0u; }
    if (t == 0) anyCen = 0u;
    __syncthreads();

    const float* m0 = mask + (size_t)b * H * W;
    const int col   = t & 63;
    const int rbase = t >> 6;

    bool colAny = false;
    bool cenAny = false;

    #pragma unroll 4
    for (int i = 0; i < 16; ++i) {
        const int r   = rbase * 16 + i;
        const int gy0 = gy * CH + r;
        const int gx0 = gx * CW + col;
        const float m = m0[(size_t)gy0 * W + gx0];
        const bool act = (m >= 0.5f);
        colAny |= act;
        if (act) atomicOr(&actRow[r], 1u);

        float nm = -__builtin_inff();
        #pragma unroll
        for (int dy = -1; dy <= 1; ++dy) {
            const int yy = gy0 + dy;
            if (yy < 0 || yy >= H) continue;
            #pragma unroll
            for (int dxn = -1; dxn <= 1; ++dxn) {
                if (dy == 0 && dxn == 0) continue;
                const int xx = gx0 + dxn;
                if (xx < 0 || xx >= W) continue;
                nm = fmaxf(nm, m0[(size_t)yy * W + xx]);
            }
        }
        cenAny |= (act && (nm <= m));
    }
    if (colAny) atomicOr(&actCol[col], 1u);
    if (cenAny) atomicOr(&anyCen, 1u);
    __syncthreads();

    if (t == 0) {
        int fc = -1, lc = -1, fr = -1, lr = -1;
        for (int i = 0; i < CW; ++i) if (actCol[i]) { if (fc < 0) fc = i; lc = i; }
        for (int i = 0; i < CH; ++i) if (actRow[i]) { if (fr < 0) fr = i; lr = i; }

        const int dx1 = (fc < 0) ? 0 : fc;
        const int dx2 = (lc < 0) ? 0 : -(63 - lc);
        const int dy1 = (fr < 0) ? 0 : fr;
        const int dy2 = (lr < 0) ? 0 : -(63 - lr);
        const int dx = (iabs_(dx1) > iabs_(dx2)) ? dx1 : dx2;
        const int dy = (iabs_(dy1) > iabs_(dy2)) ? dy1 : dy2;

        int x1 = gx * CW + dx; x1 = x1 < 0 ? 0 : (x1 > (W - CW) ? (W - CW) : x1);
        int y1 = gy * CH + dy; y1 = y1 < 0 ? 0 : (y1 > (H - CH) ? (H - CH) : y1);
        const int vld = anyCen ? 1 : 0;

        float* o = offsets_out + (size_t)n * 5;
        o[0] = (float)b;
        o[1] = (float)x1;
        o[2] = (float)y1;
        o[3] = (float)(x1 + CW);
        o[4] = (float)(y1 + CH);
        valid_out[n] = (float)vld;

        ws[3 * n + 0] = x1;
        ws[3 * n + 1] = y1;
        ws[3 * n + 2] = vld;
    }
}

// ---------------------------------------------------------------------------
// Stage 2 (TDM): one wave per (region, channel) plane. One tensor_load_to_lds
// pulls the strided 64x64 tile (row stride 512) into LDS; one
// tensor_store_from_lds pushes the packed 16KB plane to the contiguous output.
// ---------------------------------------------------------------------------
__global__ void __launch_bounds__(32)
patch_stage2_tdm(const float* __restrict__ x,
                 const int* __restrict__ ws,
                 float* __restrict__ out)
{
    __shared__ float smem[CH * CW];    // 16KB tile

    const int c = blockIdx.x;          // channel 0..63
    const int n = blockIdx.y;          // region 0..255
    const int b = n >> 6;
    const int t = threadIdx.x;         // 32 threads = 1 wave

    const int x1  = ws[3 * n + 0];
    const int y1  = ws[3 * n + 1];
    const int vld = ws[3 * n + 2];

    float* dst_plane = out + (((size_t)n * C + c) << 12);   // * 64*64

    if (!vld) {
        float4 z = make_float4(0.f, 0.f, 0.f, 0.f);
        float4* d4 = (float4*)dst_plane;                    // 1024 float4
        #pragma unroll
        for (int i = 0; i < 32; ++i) d4[t + 32 * i] = z;
        return;
    }

    const float* src = x + (((size_t)b * C + c) << 18)      // * 512*512
                         + (size_t)y1 * W + x1;             // tile origin

    const uint32_t lds = (uint32_t)(uintptr_t)(&smem[0]);

    // Load: 64x64 tile of a 512x512 (row stride 512) f32 tensor -> LDS packed.
    u32x4 g0l = make_g0(lds, (uint64_t)(uintptr_t)src);
    u32x8 g1l = make_g1(/*tdim0=*/W, /*tdim1=*/H,
                        /*tile0=*/CW, /*tile1=*/CH, /*stride0=*/W);
    // Store: LDS packed 4096 floats -> contiguous plane (1 row of 4096).
    u32x4 g0s = make_g0(lds, (uint64_t)(uintptr_t)dst_plane);
    u32x8 g1s = make_g1(/*tdim0=*/CH * CW, /*tdim1=*/1,
                        /*tile0=*/CH * CW, /*tile1=*/1, /*stride0=*/CH * CW);

    tdm_load(g0l, g1l);
    __builtin_amdgcn_s_wait_tensorcnt(0);    // tile resident in LDS
    tdm_store(g0s, g1s);
    __builtin_amdgcn_s_wait_tensorcnt(0);    // store done before LDS freed
}

// ---------------------------------------------------------------------------
// Stage 2 (ASYNCcnt DMA) — alternate path, kept compiled.
// ---------------------------------------------------------------------------
__global__ void patch_stage2_async(const float* __restrict__ x,
                                   const int* __restrict__ ws,
                                   float* __restrict__ out)
{
    __shared__ float smem[4096];   // 16KB: 8 waves * 2KB

    const int c = blockIdx.x;
    const int n = blockIdx.y;
    const int b = n >> 6;
    const int t = threadIdx.x;

    const int x1  = ws[3 * n + 0];
    const int y1  = ws[3 * n + 1];
    const int vld = ws[3 * n + 2];

    float* dst_plane = out + (((size_t)n * C + c) << 12);

    if (!vld) {
        float4 z = make_float4(0.f, 0.f, 0.f, 0.f);
        float4* d4 = (float4*)dst_plane;
        #pragma unroll
        for (int i = 0; i < 4; ++i) d4[t + 256 * i] = z;
        return;
    }

    const int wv = t >> 5, lane = t & 31;
    const float* src_plane = x + (((size_t)b * C + c) << 18);
    const float* srow0 = src_plane + (size_t)(y1 + wv * 8) * W + x1;
    float*       drow0 = dst_plane + wv * 8 * CW;

    uint32_t ldsBase = (uint32_t)(uintptr_t)(&smem[0]) + (uint32_t)(wv * 2048);

    #pragma unroll
    for (int i = 0; i < 16; ++i) {
        const int e = lane + 32 * i;
        const int row8 = e >> 6;
        const int cw   = e & 63;
        async_load_b32(ldsBase + (uint32_t)(i * 128 + lane * 4),
                       srow0 + (size_t)row8 * W + cw);
    }
    wait_async0();

    #pragma unroll
    for (int i = 0; i < 16; ++i) {
        const int e = lane + 32 * i;
        const int row8 = e >> 6;
        const int cw   = e & 63;
        async_store_b32(drow0 + row8 * CW + cw,
                        ldsBase + (uint32_t)(i * 128 + lane * 4));
    }
    wait_async0();
}

// ---------------------------------------------------------------------------
extern "C" void kernel_launch(void* const* d_in, const int* in_sizes, int n_in,
                              void* d_out, int out_size, void* d_ws, size_t ws_size,
                              hipStream_t stream) {
    const float* x    = (const float*)d_in[0];   // (4,64,512,512) f32
    const float* heat = (const float*)d_in[1];   // (4,1,512,512) f32
    float* out = (float*)d_out;                  // patches | offsets | valid
    int*   ws  = (int*)d_ws;

    float* offsets_out = out + PATCH_ELEMS;
    float* valid_out   = offsets_out + (size_t)NREG * 5;

    heatmap_stage1  <<<dim3(NREG),    dim3(256), 0, stream>>>(heat, offsets_out, valid_out, ws);
    patch_stage2_tdm<<<dim3(C, NREG), dim3(32),  0, stream>>>(x, ws, out);
}